// EM_rec_loss_32555852104027
// MI455X (gfx1250) — compile-verified
//
#include <hip/hip_runtime.h>
#include <hip/hip_bf16.h>

// Problem constants (from reference)
#define B_   8
#define F_   6
#define NB_  5
#define NS_  7
#define H_   192
#define W_   192
#define C_   3
#define HW_  (H_*W_)
#define PLANE_B  (HW_*4)        // bytes per plane = 147456 (fits 24-bit ioffset)
#define NBF  (B_*(F_-1))        // 40 (b, frame) pairs, frames 1..5
#define CHUNK   1024            // pixels per block
#define NCHUNK  (HW_/CHUNK)     // 36
#define ACCSZ   80              // per-(b,f): 35 ce_pos + 35 mse + 5 base (+5 pad)
#define LN2_F 0.69314718055994530942f

typedef __attribute__((ext_vector_type(2))) float f32x2;
typedef __attribute__((ext_vector_type(8))) float f32x8;

#define LD2(ptr, byteoff) (*(const f32x2*)((const char*)(ptr) + (byteoff)))

__global__ void em_init(float* __restrict__ ws) {
    int i = blockIdx.x * blockDim.x + threadIdx.x;
    if (i < NBF * ACCSZ) ws[i] = 0.0f;
}

// Streaming reduction: one (b,f) pair x 1024-pixel chunk per block.
// Each wave accumulates one 16x16 fp32 V_WMMA_F32_16X16X4_F32 tile:
//   A rows 0-4: logp-log1mp (buffer n), 5-9: sum_c(rec-tgt)^2, 10-14: log1mp
//   B cols 0-6: t_s (mask>0.5), 7-13: vis_s (vis>0.5), 14: 1.0
// -> C[0:5,0:7]=ce_pos, C[5:10,7:14]=mse, C[10:15,14]=ce_base. Row/col 15 junk.
__global__ __launch_bounds__(256) void em_main(
    const float* __restrict__ seg_g, const float* __restrict__ msk_g,
    const float* __restrict__ rec_g, const float* __restrict__ tgt_g,
    const float* __restrict__ vis_g, float* __restrict__ ws)
{
    __shared__ float sAcc[ACCSZ];
    if (threadIdx.x < ACCSZ) sAcc[threadIdx.x] = 0.0f;
    __syncthreads();

    const int bf    = blockIdx.x / NCHUNK;
    const int chunk = blockIdx.x % NCHUNK;
    const int b  = bf / (F_ - 1);
    const int f1 = bf % (F_ - 1) + 1;          // frames 1..5

    const float* seg = seg_g + (size_t)((b*F_ + f1)*NB_)      * HW_;
    const float* msk = msk_g + (size_t)((b*F_ + f1)*NS_)      * HW_;
    const float* rec = rec_g + (size_t)((b*F_ + f1)*NB_) * C_ * HW_;
    const float* tgt = tgt_g + (size_t)((b*F_ + f1)*C_)       * HW_;
    const float* vis = vis_g + (size_t)((b*F_ + f1)*NS_)      * HW_;

    const int lane = threadIdx.x & 31;
    const int wave = threadIdx.x >> 5;
    const int half = lane >> 4;                // K-pair selector (ISA A/B layout)
    const int role = lane & 15;                // A-row / B-col index

    // ---- per-lane A stream: one base pointer; recon/tgt c-planes via ioffset ----
    const bool eRole = (role >= 5 && role < 10);   // rows 5-9: squared-error rows
    const bool dRole = (role < 5);                 // rows 0-4: logp-log1mp
    const float* pA0;
    if (dRole)          pA0 = seg + (size_t)role * HW_;
    else if (eRole)     pA0 = rec + (size_t)((role - 5)*C_) * HW_;  // c0 plane
    else                pA0 = seg + (size_t)((role < 15) ? (role - 10) : 0) * HW_; // 15: dummy
    // log-role FMA coefficients: av = log2(x)*cA + log2(1-x)*cB  (natural-log folded)
    const float cA = dRole ? LN2_F : 0.0f;
    const float cB = (role >= 15) ? 0.0f : (dRole ? -LN2_F : LN2_F);

    // ---- per-lane B stream: always load; per-lane threshold folds constants ----
    //  role<14: >0.5 threshold; role 14: always-true -> 1.0; role 15: never -> 0.0
    const float bThr = (role < 14) ? 0.5f : ((role == 14) ? -3.0e38f : 3.0e38f);
    const float* pB = (role < 7)  ? (msk + (size_t)role * HW_)
                    : (role < 14) ? (vis + (size_t)(role - 7) * HW_)
                                  : msk;       // dummy stream for 14/15

    const int base = chunk * CHUNK + wave * (CHUNK / 8) + 2 * half;
    const char* qa = (const char*)(pA0 + base);
    const char* qt = (const char*)(tgt + base);
    const char* qb = (const char*)(pB  + base);

    const float EPS = 1e-7f;
    const float OME = 1.0f - 1e-7f;
    f32x8 acc = {0.f, 0.f, 0.f, 0.f, 0.f, 0.f, 0.f, 0.f};

    for (int g = 0; g < 8; ++g) {              // 8 groups x 16 px = 128 px per wave
        __builtin_prefetch(qa + 512, 0, 0);    // global_prefetch_b8 -> GL2
        __builtin_prefetch(qb + 512, 0, 0);

        float av[8];                           // a-values: 4 steps x 2 K-slots
        if (!eRole) {
            // rows 0-4 and 10-14 execute these instructions under one EXEC mask
            #pragma unroll
            for (int u = 0; u < 4; ++u) {
                f32x2 x = LD2(qa, 16*u);
                float x0 = __builtin_fminf(__builtin_fmaxf(x.x, EPS), OME); // v_med3
                float x1 = __builtin_fminf(__builtin_fmaxf(x.y, EPS), OME);
                float g0 = __builtin_amdgcn_logf(x0);          // raw v_log_f32 (log2)
                float g1 = __builtin_amdgcn_logf(x1);
                float h0 = __builtin_amdgcn_logf(1.0f - x0);   // log2(1-x), 1-x exact-ish
                float h1 = __builtin_amdgcn_logf(1.0f - x1);
                av[2*u+0] = __builtin_fmaf(g0, cA, h0 * cB);
                av[2*u+1] = __builtin_fmaf(g1, cA, h1 * cB);
            }
        } else {
            #pragma unroll
            for (int u = 0; u < 4; ++u) {
                f32x2 r0 = LD2(qa, 16*u);
                f32x2 r1 = LD2(qa, 16*u +   PLANE_B);
                f32x2 r2 = LD2(qa, 16*u + 2*PLANE_B);
                f32x2 t0 = LD2(qt, 16*u);
                f32x2 t1 = LD2(qt, 16*u +   PLANE_B);
                f32x2 t2 = LD2(qt, 16*u + 2*PLANE_B);
                float d0 = r0.x - t0.x, d1 = r1.x - t1.x, d2 = r2.x - t2.x;
                float e0 = r0.y - t0.y, e1 = r1.y - t1.y, e2 = r2.y - t2.y;
                av[2*u+0] = __builtin_fmaf(d0, d0, __builtin_fmaf(d1, d1, d2*d2));
                av[2*u+1] = __builtin_fmaf(e0, e0, __builtin_fmaf(e1, e1, e2*e2));
            }
        }

        // Branch-free B construction + 4 WMMAs at full EXEC
        #pragma unroll
        for (int u = 0; u < 4; ++u) {
            f32x2 mv = LD2(qb, 16*u);          // roles 14/15 read dummy (cached) line
            float b0 = (mv.x > bThr) ? 1.0f : 0.0f;   // cmp + cndmask only
            float b1 = (mv.y > bThr) ? 1.0f : 0.0f;
            f32x2 Av = {av[2*u+0], av[2*u+1]};
            f32x2 Bv = {b0, b1};
            acc = __builtin_amdgcn_wmma_f32_16x16x4_f32(
                false, Av, false, Bv, (short)0, acc, false, false);
        }

        qa += 64; qt += 64; qb += 64;          // 16 px * 4 B
    }

    // C layout: VGPR j -> row M = j + 8*half, col N = role (ds_add_f32 scatter)
    #pragma unroll
    for (int j = 0; j < 8; ++j) {
        int M = j + (half << 3);
        int N = role;
        float v = acc[j];
        int idx = -1;
        if (M < 5 && N < 7)                            idx = M*7 + N;              // ce_pos
        else if (M >= 5 && M < 10 && N >= 7 && N < 14) idx = 35 + (M-5)*7 + (N-7); // mse
        else if (M >= 10 && M < 15 && N == 14)         idx = 70 + (M-10);          // ce_base
        if (idx >= 0) atomicAdd(&sAcc[idx], v);
    }
    __syncthreads();
    if (threadIdx.x < ACCSZ)
        atomicAdd(&ws[bf*ACCSZ + threadIdx.x], sAcc[threadIdx.x]);
}

__global__ __launch_bounds__(256) void em_finalize(
    const float* __restrict__ ws, const float* __restrict__ attn,
    float* __restrict__ out)
{
    __shared__ float red[256];
    float sum = 0.0f;
    for (int idx = threadIdx.x; idx < NBF * NB_ * NS_; idx += blockDim.x) {
        int bf = idx / (NB_*NS_);
        int r  = idx % (NB_*NS_);
        int n  = r / NS_;
        int s  = r % NS_;
        int b  = bf / (F_-1);
        int f1 = bf % (F_-1) + 1;
        const float* a = ws + bf * ACCSZ;
        float cepos = a[n*NS_ + s];
        float msev  = a[35 + n*NS_ + s];
        float basev = a[70 + n];
        float bce   = -(cepos + basev) * (1.0f / (float)HW_);
        float loss  = bce + 0.1f * (msev * (1.0f / (float)C_));
        float w     = attn[(((size_t)b*F_ + f1)*NS_ + s)*NB_ + n];
        sum += w * loss;
    }
    red[threadIdx.x] = sum;
    __syncthreads();
    for (int off = 128; off > 0; off >>= 1) {
        if (threadIdx.x < off) red[threadIdx.x] += red[threadIdx.x + off];
        __syncthreads();
    }
    if (threadIdx.x == 0)
        out[0] = red[0] * (20.0f / (float)(B_*(F_-1)*NB_*NS_));
}

extern "C" void kernel_launch(void* const* d_in, const int* in_sizes, int n_in,
                              void* d_out, int out_size, void* d_ws, size_t ws_size,
                              hipStream_t stream) {
    (void)in_sizes; (void)n_in; (void)out_size; (void)ws_size;
    const float* seg  = (const float*)d_in[0];
    const float* msk  = (const float*)d_in[1];
    const float* rec  = (const float*)d_in[2];
    const float* tgt  = (const float*)d_in[3];
    const float* vis  = (const float*)d_in[4];
    const float* attn = (const float*)d_in[5];
    float* out = (float*)d_out;
    float* ws  = (float*)d_ws;

    em_init<<<(NBF*ACCSZ + 255)/256, 256, 0, stream>>>(ws);
    em_main<<<NBF * NCHUNK, 256, 0, stream>>>(seg, msk, rec, tgt, vis, ws);
    em_finalize<<<1, 256, 0, stream>>>(ws, attn, out);
}